// NNConv_14044543058374
// MI455X (gfx1250) — compile-verified
//
#include <hip/hip_runtime.h>
#include <hip/hip_bf16.h>

#define N_NODES   50000
#define N_ROWS_PAD 50048       // padded row count for P (covers 391*128 grid rows)
#define N_EDGES   200000
#define IN_FEATS  32
#define OUT_FEATS 32
#define EDGE_DIM  16
#define NCOLS     544          // 16*32 edge-weight cols + 32 bias cols
#define HALF_T    17           // 17 tiles of 16 cols per LDS staging (272 cols)
#define LDK       33           // padded K stride in LDS (bank-conflict avoidance)

typedef __attribute__((ext_vector_type(2))) float v2f;
typedef __attribute__((ext_vector_type(8))) float v8f;

// ---------------------------------------------------------------------------
// Kernel 1: zero the atomic accumulators (aggsum + deg) in workspace.
// ---------------------------------------------------------------------------
__global__ void nnconv_zero(float* __restrict__ p, int n) {
    int i = blockIdx.x * blockDim.x + threadIdx.x;
    if (i < n) p[i] = 0.0f;
}

// ---------------------------------------------------------------------------
// Kernel 2: P[m, c] = sum_i feat[m,i] * W2[i,c]
//   c in [0,512):  W2[i, d*32+o] = W_edge[d*1024 + i*32 + o]
//   c in [512,544): W2[i, 512+o] = b_edge[i*32 + o]
// GEMM M=50000(padded 50048) K=32 N=544 via V_WMMA_F32_16X16X4_F32.
// One wave = one 16-row M tile; block = 8 waves = 128 rows.
// P is padded so D stores need no bounds check (tail rows are scratch-only).
// ---------------------------------------------------------------------------
__global__ __launch_bounds__(256) void nnconv_node_transform(
    const float* __restrict__ feat,
    const float* __restrict__ W_edge,
    const float* __restrict__ b_edge,
    float* __restrict__ P)
{
    __shared__ float Bs[HALF_T * 16 * LDK];   // 272 cols * 33 * 4B = 35,904 B

    const int tid  = threadIdx.x;
    const int wave = tid >> 5;
    const int lane = tid & 31;
    const int half = lane >> 4;     // 0: lanes 0-15 (K 0,1) | 1: lanes 16-31 (K 2,3)
    const int lm   = lane & 15;

    const int mbase = blockIdx.x * 128 + wave * 16;
    const int row   = mbase + lm;
    const int rowc  = (row < N_NODES) ? row : 0;   // clamp input loads only

    // A fragments for the full K=32: a[k] supplies A[M=lm, K=4k+2*half+{0,1}]
    v2f a[8];
    const float* frow = feat + (size_t)rowc * IN_FEATS + 2 * half;
    #pragma unroll
    for (int k = 0; k < 8; ++k) {
        a[k].x = frow[4 * k + 0];
        a[k].y = frow[4 * k + 1];
    }

    for (int hblk = 0; hblk < 2; ++hblk) {
        const int cbase = hblk * (HALF_T * 16);

        __syncthreads();   // previous staging fully consumed
        // Stage B half: Bs[cl*LDK + i] = W2[i, cbase+cl]
        for (int idx = tid; idx < HALF_T * 16 * IN_FEATS; idx += 256) {
            int cl = idx >> 5;          // local col 0..271
            int i  = idx & 31;          // K index 0..31
            int c  = cbase + cl;
            float v;
            if (c < 512) {
                int d = c >> 5, o = c & 31;
                v = W_edge[d * 1024 + i * 32 + o];
            } else {
                v = b_edge[i * 32 + (c - 512)];
            }
            Bs[cl * LDK + i] = v;
        }
        __syncthreads();

        for (int nt = 0; nt < HALF_T; ++nt) {
            const int cl = nt * 16 + lm;                 // this lane's B column N=lm
            const float* bcol = &Bs[cl * LDK + 2 * half];
            v8f acc = {};
            #pragma unroll
            for (int k = 0; k < 8; ++k) {
                v2f b;                                   // B[K=4k+2*half+{0,1}, N=lm]
                b.x = bcol[4 * k + 0];
                b.y = bcol[4 * k + 1];
                acc = __builtin_amdgcn_wmma_f32_16x16x4_f32(
                    false, a[k], false, b, (short)0, acc, false, false);
            }
            // D layout: lane holds col N=lm, VGPR r holds row M = r + 8*half.
            // P rows are padded to N_ROWS_PAD -> unconditional stores.
            const int c = cbase + nt * 16 + lm;
            float* pw = P + (size_t)(mbase + 8 * half) * NCOLS + c;
            #pragma unroll
            for (int r = 0; r < 8; ++r) {
                pw[(size_t)r * NCOLS] = acc[r];
            }
        }
    }
}

// ---------------------------------------------------------------------------
// Kernel 3: one wave per edge, lane = output feature o.
//   msg[o] = P[src, 512+o] + sum_d efeat[e,d] * P[src, d*32+o]
//   atomic segment-sum into aggsum[dst, o]; degree count.
// ---------------------------------------------------------------------------
__global__ __launch_bounds__(256) void nnconv_edge_msg(
    const float* __restrict__ P,
    const float* __restrict__ efeat,
    const int*   __restrict__ src,
    const int*   __restrict__ dst,
    float* __restrict__ aggsum,
    float* __restrict__ deg)
{
    const int e = blockIdx.x * 8 + (threadIdx.x >> 5);
    if (e >= N_EDGES) return;
    const int lane = threadIdx.x & 31;

    const int s = __builtin_amdgcn_readfirstlane(src[e]);
    const int d = __builtin_amdgcn_readfirstlane(dst[e]);

    const float* prow = P + (size_t)s * NCOLS;
    float acc = prow[512 + lane];                       // bias-block term (h . Bre)

    // efeat row broadcast: lanes 0-15 carry the 16 edge features
    float ew = efeat[(size_t)e * EDGE_DIM + (lane & 15)];
    #pragma unroll
    for (int dd = 0; dd < EDGE_DIM; ++dd) {
        float wd = __shfl(ew, dd, 32);
        acc += wd * prow[dd * 32 + lane];               // coalesced 128B per dd
    }

    atomicAdd(&aggsum[(size_t)d * OUT_FEATS + lane], acc);
    if (lane == 0) atomicAdd(&deg[d], 1.0f);
}

// ---------------------------------------------------------------------------
// Kernel 4: out[n,o] = aggsum[n,o] / max(deg[n],1) + bias[o]
// ---------------------------------------------------------------------------
__global__ void nnconv_finalize(
    const float* __restrict__ aggsum,
    const float* __restrict__ deg,
    const float* __restrict__ bias,
    float* __restrict__ out)
{
    int i = blockIdx.x * blockDim.x + threadIdx.x;
    if (i < N_NODES * OUT_FEATS) {
        int n = i >> 5, o = i & 31;
        float dg = deg[n];
        out[i] = aggsum[i] / fmaxf(dg, 1.0f) + bias[o];
    }
}

// ---------------------------------------------------------------------------
extern "C" void kernel_launch(void* const* d_in, const int* in_sizes, int n_in,
                              void* d_out, int out_size, void* d_ws, size_t ws_size,
                              hipStream_t stream) {
    const float* feat   = (const float*)d_in[0];   // (50000, 32)
    const float* efeat  = (const float*)d_in[1];   // (200000, 16)
    const int*   src    = (const int*)  d_in[2];   // (200000,)
    const int*   dst    = (const int*)  d_in[3];   // (200000,)
    const float* W_edge = (const float*)d_in[4];   // (16, 1024)
    const float* b_edge = (const float*)d_in[5];   // (1024,)
    const float* bias   = (const float*)d_in[6];   // (32,)
    float* out = (float*)d_out;                    // (50000, 32)

    // Workspace layout (bytes):
    //   P      @ 0            : 50048*544*4 = 108,904,448  (rows padded)
    //   aggsum @ 108,904,448  : 50000*32*4  =   6,400,000
    //   deg    @ 115,304,448  : 50000*4     =     200,000
    char* ws = (char*)d_ws;
    float* P      = (float*)(ws);
    float* aggsum = (float*)(ws + 108904448u);
    float* degbuf = (float*)(ws + 115304448u);

    // 1) zero aggsum+deg (contiguous: 1,650,000 floats)
    {
        int n = N_NODES * OUT_FEATS + N_NODES;
        nnconv_zero<<<(n + 255) / 256, 256, 0, stream>>>(aggsum, n);
    }
    // 2) per-node transform GEMM (WMMA f32 16x16x4)
    {
        int grid = (N_NODES + 127) / 128;   // 391 blocks; padded P absorbs tail rows
        nnconv_node_transform<<<grid, 256, 0, stream>>>(feat, W_edge, b_edge, P);
    }
    // 3) per-edge contraction + atomic segment-sum
    {
        int grid = (N_EDGES + 7) / 8;
        nnconv_edge_msg<<<grid, 256, 0, stream>>>(P, efeat, src, dst, aggsum, degbuf);
    }
    // 4) mean + bias
    {
        int n = N_NODES * OUT_FEATS;
        nnconv_finalize<<<(n + 255) / 256, 256, 0, stream>>>(aggsum, degbuf, bias, out);
    }
}